// Deformer_43688407335001
// MI455X (gfx1250) — compile-verified
//
#include <hip/hip_runtime.h>
#include <math.h>

// Problem constants (match reference)
#define Hh 192
#define Ww 192
#define Nb 6
#define KC 400           // control points (20x20)
#define KA 403           // + affine rows
#define AW 408           // A row stride (403 cols + pad + 2 rhs + pad)
#define HWp (Hh*Ww)

typedef float v2f __attribute__((ext_vector_type(2)));
typedef float v8f __attribute__((ext_vector_type(8)));

#define LN2F 0.69314718055994530942f

// Full-precision U for the system build (feeds the solve).
__device__ __forceinline__ float tpsU(float r2) {
    return 0.5f * r2 * __logf(r2 + 1e-9f);
}
// Fast U for the apply loop: argument is >= 1e-9 (never denormal/negative),
// so raw v_log_f32 (log2) * ln2 is safe and skips the slow-path corrections.
__device__ __forceinline__ float tpsU_fast(float r2) {
    return 0.5f * r2 * (__builtin_amdgcn_logf(r2 + 1e-9f) * LN2F);
}

// ---------------------------------------------------------------------------
// Stage 1: backward bilinear warp of delta/coord/mask through uv.
// One thread per (n, pixel). Also writes the thresholded mask output.
// ---------------------------------------------------------------------------
__global__ void k_warp(const float* __restrict__ duv, const float* __restrict__ uv,
                       float* __restrict__ coordW, float* __restrict__ deltaW,
                       float* __restrict__ maskOut) {
    int idx = blockIdx.x * blockDim.x + threadIdx.x;
    if (idx >= Nb * HWp) return;
    float ux = uv[(size_t)idx * 2 + 0];
    float uy = uv[(size_t)idx * 2 + 1];
    float x = (ux + 1.0f) * 0.5f * (Ww - 1);
    float y = (uy + 1.0f) * 0.5f * (Hh - 1);
    float x0 = floorf(x), y0 = floorf(y);
    float wx = x - x0, wy = y - y0;
    float cS0 = 0.f, cS1 = 0.f, dS0 = 0.f, dS1 = 0.f, mS = 0.f;
#pragma unroll
    for (int cy = 0; cy < 2; ++cy) {
#pragma unroll
        for (int cx = 0; cx < 2; ++cx) {
            float xi = x0 + (float)cx, yi = y0 + (float)cy;
            float w = (cx ? wx : 1.0f - wx) * (cy ? wy : 1.0f - wy);
            float inb = (xi >= 0.f && xi <= (float)(Ww - 1) &&
                         yi >= 0.f && yi <= (float)(Hh - 1)) ? 1.0f : 0.0f;
            int xc = (int)fminf(fmaxf(xi, 0.f), (float)(Ww - 1));
            int yc = (int)fminf(fmaxf(yi, 0.f), (float)(Hh - 1));
            float du = duv[((size_t)yc * Ww + xc) * 2 + 0];
            float dv = duv[((size_t)yc * Ww + xc) * 2 + 1];
            float val = (fabsf(du) <= 1.0f && fabsf(dv) <= 1.0f) ? 1.0f : 0.0f;
            float gx = -1.0f + 2.0f * (float)xc / (float)(Ww - 1);
            float gy = -1.0f + 2.0f * (float)yc / (float)(Hh - 1);
            float wi = w * inb;
            cS0 += wi * gx;
            cS1 += wi * gy;
            dS0 += wi * (du - gx) * val;
            dS1 += wi * (dv - gy) * val;
            mS  += wi * val;
        }
    }
    coordW[(size_t)idx * 2 + 0] = cS0;
    coordW[(size_t)idx * 2 + 1] = cS1;
    deltaW[(size_t)idx * 2 + 0] = dS0;
    deltaW[(size_t)idx * 2 + 1] = dS1;
    maskOut[idx] = (mS > 0.5f) ? 1.0f : 0.0f;
}

// ---------------------------------------------------------------------------
// Stage 2a: gather control-point coords + mask at the 20x20 lattice.
// ---------------------------------------------------------------------------
__global__ void k_ctrl(const float* __restrict__ coordW, const float* __restrict__ maskOut,
                       float* __restrict__ srcW, float* __restrict__ mctlW) {
    int idx = blockIdx.x * blockDim.x + threadIdx.x;
    if (idx >= Nb * KC) return;
    int n = idx / KC, t = idx % KC;
    int a = t / 20, b = t % 20;
    int ii = (int)rintf((float)a * (float)(Hh - 1) / 19.0f);
    int jj = (int)rintf((float)b * (float)(Ww - 1) / 19.0f);
    int pix = ii * Ww + jj;
    srcW[(size_t)idx * 2 + 0] = coordW[((size_t)n * HWp + pix) * 2 + 0];
    srcW[(size_t)idx * 2 + 1] = coordW[((size_t)n * HWp + pix) * 2 + 1];
    mctlW[idx] = maskOut[(size_t)n * HWp + pix];
}

// ---------------------------------------------------------------------------
// Stage 2b: build augmented system A (403 x 408): [K+diag, P | 0 | rhs | 0].
// One block per (n,row), threads stride the columns.
// ---------------------------------------------------------------------------
__global__ void k_build(const float* __restrict__ srcW, const float* __restrict__ mctlW,
                        float* __restrict__ A) {
    const int n = blockIdx.x / KA, row = blockIdx.x % KA;
    float* ar = A + ((size_t)n * KA + row) * AW;
    const float* src = srcW + (size_t)n * (KC * 2);
    const float* mc  = mctlW + (size_t)n * KC;
    float sx = 0.f, sy = 0.f;
    if (row < KC) { sx = src[row * 2]; sy = src[row * 2 + 1]; }
    for (int j = threadIdx.x; j < AW; j += blockDim.x) {
        float v = 0.0f;
        if (row < KC) {
            if (j < KC) {
                float dx = sx - src[j * 2], dy = sy - src[j * 2 + 1];
                v = tpsU(dx * dx + dy * dy);
                if (j == row) v += 100.0f + 1e8f * (1.0f - mc[row]);
            } else if (j == 400) v = 1.0f;
            else if (j == 401) v = sx;
            else if (j == 402) v = sy;
            else if (j == 404 || j == 405) {
                int a = row / 20, b = row % 20;
                int ii = (int)rintf((float)a * (float)(Hh - 1) / 19.0f);
                int jj = (int)rintf((float)b * (float)(Ww - 1) / 19.0f);
                v = (j == 404) ? (-1.0f + 2.0f * (float)jj / (float)(Ww - 1))
                               : (-1.0f + 2.0f * (float)ii / (float)(Hh - 1));
            }
        } else {
            if (j < KC) {
                int r = row - KC;
                v = (r == 0) ? 1.0f : ((r == 1) ? src[j * 2] : src[j * 2 + 1]);
            }
        }
        ar[j] = v;
    }
}

// ---------------------------------------------------------------------------
// Stage 2c: per-batch Gauss-Jordan elimination (no pivoting needed: the K
// block is strongly diagonally dominant from LAMBD/BIG, and the 3x3 Schur
// complement is nonsingular). sol stored [n][404][2], row 403 zero-padded.
// ---------------------------------------------------------------------------
__global__ void k_solve(float* __restrict__ A, float* __restrict__ sol) {
    const int n = blockIdx.x;
    float* a = A + (size_t)n * KA * AW;
    const int t = threadIdx.x;
    for (int k = 0; k < KA; ++k) {
        const float inv = 1.0f / a[(size_t)k * AW + k];
        for (int r = t; r < KA; r += blockDim.x) {
            if (r == k) continue;
            float f = a[(size_t)r * AW + k] * inv;
            if (f != 0.0f) {
                for (int j = k; j < 406; ++j)
                    a[(size_t)r * AW + j] -= f * a[(size_t)k * AW + j];
            }
        }
        __threadfence();
        __syncthreads();
    }
    for (int r = t; r < 404; r += blockDim.x) {
        if (r < KA) {
            float d = a[(size_t)r * AW + r];
            sol[((size_t)n * 404 + r) * 2 + 0] = a[(size_t)r * AW + 404] / d;
            sol[((size_t)n * 404 + r) * 2 + 1] = a[(size_t)r * AW + 405] / d;
        } else {
            sol[((size_t)n * 404 + r) * 2 + 0] = 0.0f;
            sol[((size_t)n * 404 + r) * 2 + 1] = 0.0f;
        }
    }
}

// ---------------------------------------------------------------------------
// Stage 3: fused TPS apply (WMMA f32 16x16x4) for the 3 query sets, finite-
// difference Jacobian, and final deform assembly.
//
// Per wave: 16 query pixels (M). K = 404 (400 radial + [1,x,y] + pad) chunked
// into 101 slices of 4. Chunk loop is OUTER so the B fragment (weights) and
// the control-point pair are fetched once and reused by all 3 query sets
// (3 live f32 accumulators). B loads are unconditional (clamped LDS address
// + v_cndmask) to avoid exec-masked divergent DS loads.
//
// A fragment (16x4 f32, 2 VGPRs): lanes 0-15 hold K=0,1; lanes 16-31 K=2,3.
// C/D (16x16 f32, 8 VGPRs): M = vgpr + 8*(lane>=16), N = lane&15.
// ---------------------------------------------------------------------------
__global__ void __launch_bounds__(256) k_tps_apply(
        const float* __restrict__ coordW, const float* __restrict__ deltaW,
        const float* __restrict__ srcW, const float* __restrict__ solW,
        const float* __restrict__ maskOut, float* __restrict__ deformOut) {
    __shared__ float solS[404 * 2];
    __shared__ float srcS[KC * 2];
    __shared__ float stg[8][3][16][2];

    const int bpb = HWp / 128;               // blocks per batch = 288
    const int n = blockIdx.x / bpb;
    const int tid = threadIdx.x;
    const int lane = tid & 31, wid = tid >> 5;
    const int mrow = lane & 15;              // M row (and N col for C/D)
    const int half = lane >> 4;              // K half for A/B fragments

    for (int t = tid; t < 404 * 2; t += 256) solS[t] = solW[(size_t)n * 808 + t];
    for (int t = tid; t < KC * 2; t += 256)  srcS[t] = srcW[(size_t)n * 800 + t];
    __syncthreads();

    const int p = (blockIdx.x % bpb) * 128 + wid * 16 + mrow;
    const float qx = coordW[((size_t)n * HWp + p) * 2 + 0];
    const float qy = coordW[((size_t)n * HWp + p) * 2 + 1];
    const float DX = 2.0f / (float)(Ww - 1);
    const float DY = 2.0f / (float)(Hh - 1);

    const float bmask = (mrow < 2) ? 1.0f : 0.0f;
    const int   bcol  = mrow & 1;            // clamped column for safe LDS read

    v8f acc[3];
#pragma unroll
    for (int s = 0; s < 3; ++s) acc[s] = (v8f){0.f,0.f,0.f,0.f,0.f,0.f,0.f,0.f};

    for (int chunk = 0; chunk < 101; ++chunk) {
        const int klo = chunk * 4 + half * 2;
        const bool radial = (klo < KC);
        const int kidx = radial ? klo : 0;   // clamp to keep LDS reads in-range
        // Control-point pair (shared by all 3 sets)
        const float s0x = srcS[kidx * 2 + 0], s0y = srcS[kidx * 2 + 1];
        const float s1x = srcS[kidx * 2 + 2], s1y = srcS[kidx * 2 + 3];
        // B fragment (weights, shared by all 3 sets); row 403 is zero padding.
        v2f b;
        b.x = solS[klo * 2 + bcol] * bmask;
        b.y = solS[(klo + 1) * 2 + bcol] * bmask;

#pragma unroll
        for (int s = 0; s < 3; ++s) {
            const float qsx = qx + ((s == 1) ? DX : 0.0f);
            const float qsy = qy + ((s == 2) ? DY : 0.0f);
            v2f a;
            if (radial) {
                float d0x = qsx - s0x, d0y = qsy - s0y;
                float d1x = qsx - s1x, d1y = qsy - s1y;
                a.x = tpsU_fast(d0x * d0x + d0y * d0y);
                a.y = tpsU_fast(d1x * d1x + d1y * d1y);
            } else if (klo == 400) {         // affine columns [1, x] (lanes 0-15)
                a.x = 1.0f; a.y = qsx;
            } else {                         // [y, pad] (lanes 16-31)
                a.x = qsy; a.y = 0.0f;
            }
            acc[s] = __builtin_amdgcn_wmma_f32_16x16x4_f32(
                false, a, false, b, (short)0, acc[s], false, false);
        }
    }

#pragma unroll
    for (int s = 0; s < 3; ++s) {
        if (mrow < 2) {
#pragma unroll
            for (int v = 0; v < 8; ++v)
                stg[wid][s][v + half * 8][mrow] = acc[s][v];
        }
    }
    __syncthreads();

    if (half == 0) {
        float c0x = stg[wid][0][mrow][0], c0y = stg[wid][0][mrow][1];
        float cxx = stg[wid][1][mrow][0], cxy = stg[wid][1][mrow][1];
        float cyx = stg[wid][2][mrow][0], cyy = stg[wid][2][mrow][1];
        float aJ = (cxx - c0x) / DX;
        float bJ = (cxy - c0y) / DY;
        float cJ = (cyx - c0x) / DX;
        float dJ = (cyy - c0y) / DY;
        float dfx = deltaW[((size_t)n * HWp + p) * 2 + 0];
        float dfy = deltaW[((size_t)n * HWp + p) * 2 + 1];
        float dnx = aJ * dfx + cJ * dfy;
        float dny = bJ * dfx + dJ * dfy;
        float m = maskOut[(size_t)n * HWp + p];
        int i = p / Ww, j = p % Ww;
        float gxv = -1.0f + 2.0f * (float)j / (float)(Ww - 1);
        float gyv = -1.0f + 2.0f * (float)i / (float)(Hh - 1);
        deformOut[((size_t)n * HWp + p) * 2 + 0] = (gxv + dnx) * m - 2.0f * (1.0f - m);
        deformOut[((size_t)n * HWp + p) * 2 + 1] = (gyv + dny) * m - 2.0f * (1.0f - m);
    }
}

// ---------------------------------------------------------------------------
// Workspace layout (floats):
//   coordW : Nb*HW*2           = 442368   @ 0
//   deltaW : Nb*HW*2           = 442368   @ 442368
//   srcW   : Nb*400*2          = 4800     @ 884736
//   mctlW  : Nb*400            = 2400     @ 889536
//   solW   : Nb*404*2          = 4848     @ 891936
//   A      : Nb*403*408        = 986544   @ 896784
// total ~7.5 MB
// ---------------------------------------------------------------------------
extern "C" void kernel_launch(void* const* d_in, const int* in_sizes, int n_in,
                              void* d_out, int out_size, void* d_ws, size_t ws_size,
                              hipStream_t stream) {
    (void)in_sizes; (void)n_in; (void)out_size; (void)ws_size;
    const float* duv = (const float*)d_in[0];   // deform_uv (1,H,W,2)
    const float* uv  = (const float*)d_in[1];   // uv (N,H,W,2)
    float* out = (float*)d_out;
    float* ws  = (float*)d_ws;

    float* coordW = ws;
    float* deltaW = ws + 442368;
    float* srcW   = ws + 884736;
    float* mctlW  = ws + 889536;
    float* solW   = ws + 891936;
    float* Amat   = ws + 896784;

    float* deformOut = out;                        // (N,H,W,2)
    float* maskOut   = out + (size_t)Nb * HWp * 2; // (N,1,H,W)

    k_warp <<<(Nb * HWp + 255) / 256, 256, 0, stream>>>(duv, uv, coordW, deltaW, maskOut);
    k_ctrl <<<(Nb * KC + 255) / 256, 256, 0, stream>>>(coordW, maskOut, srcW, mctlW);
    k_build<<<Nb * KA, 256, 0, stream>>>(srcW, mctlW, Amat);
    k_solve<<<Nb, 256, 0, stream>>>(Amat, solW);
    k_tps_apply<<<Nb * (HWp / 128), 256, 0, stream>>>(coordW, deltaW, srcW, solW, maskOut, deformOut);
}